// MultiscaleDiscriminator_62457414419226
// MI455X (gfx1250) — compile-verified
//
#include <hip/hip_runtime.h>
#include <hip/hip_bf16.h>

// Problem constants (from reference file)
#define NN 100000
#define EE 1600000
#define HH 64
#define SS 3
#define GG 50

typedef __attribute__((ext_vector_type(2))) float v2f;
typedef __attribute__((ext_vector_type(8))) float v8f;

// ---------------------------------------------------------------------------
// Generic N x 64 = (N x 64) @ (64 x 64) GEMM using V_WMMA_F32_16X16X4_F32.
// One wave computes a 16x16 tile of C; a 256-thread block (8 wave32 waves)
// covers 32 rows x 64 cols. K is looped in steps of 4.
//
// VGPR layouts per CDNA5 ISA 7.12.2:
//   A (16x4 f32, 2 VGPRs): lanes 0-15 rows M=0..15; v0={K0|K2}, v1={K1|K3}
//   B (4x16 f32, 2 VGPRs): lanes 0-15 cols N=0..15; v0={K0|K2}, v1={K1|K3}
//   C/D (16x16 f32, 8 VGPRs): vgpr v -> row M = v + 8*(lane>=16), col = lane&15
// ---------------------------------------------------------------------------
template <bool RELU, bool DUAL>
__global__ __launch_bounds__(256) void gemm64_wmma(const float* __restrict__ X1,
                                                   const float* __restrict__ W1,
                                                   const float* __restrict__ X2,
                                                   const float* __restrict__ W2,
                                                   const float* __restrict__ bias,
                                                   float* __restrict__ C, int M) {
  const int wave = threadIdx.x >> 5;
  const int lane = threadIdx.x & 31;
  const int rowTile = blockIdx.x * 2 + (wave >> 2);
  const int colTile = wave & 3;
  const int row0 = rowTile * 16;
  const int col0 = colTile * 16;
  if (row0 >= M) return;

  const int half = lane >> 4;   // 0: lanes 0-15, 1: lanes 16-31
  const int lid  = lane & 15;

  v8f acc = {};

#if __has_builtin(__builtin_amdgcn_wmma_f32_16x16x4_f32)
  #pragma unroll
  for (int k0 = 0; k0 < 64; k0 += 4) {
    const int kb = k0 + half * 2;
    v2f a = *(const v2f*)(X1 + (size_t)(row0 + lid) * 64 + kb);
    v2f b;
    b.x = W1[(size_t)kb * 64 + col0 + lid];
    b.y = W1[(size_t)(kb + 1) * 64 + col0 + lid];
    acc = __builtin_amdgcn_wmma_f32_16x16x4_f32(false, a, false, b,
                                                (short)0, acc, false, false);
  }
  if (DUAL) {
    #pragma unroll
    for (int k0 = 0; k0 < 64; k0 += 4) {
      const int kb = k0 + half * 2;
      v2f a = *(const v2f*)(X2 + (size_t)(row0 + lid) * 64 + kb);
      v2f b;
      b.x = W2[(size_t)kb * 64 + col0 + lid];
      b.y = W2[(size_t)(kb + 1) * 64 + col0 + lid];
      acc = __builtin_amdgcn_wmma_f32_16x16x4_f32(false, a, false, b,
                                                  (short)0, acc, false, false);
    }
  }
  const float bv = bias ? bias[col0 + lid] : 0.0f;
  #pragma unroll
  for (int v = 0; v < 8; ++v) {
    const int r = row0 + v + 8 * half;
    float val = acc[v] + bv;
    if (RELU) val = fmaxf(val, 0.0f);
    C[(size_t)r * 64 + col0 + lid] = val;
  }
#else
  // Scalar fallback (should not be taken on gfx1250; histogram would show wmma=0)
  const float bv = bias ? bias[col0 + lid] : 0.0f;
  for (int v = 0; v < 8; ++v) {
    const int r = row0 + v + 8 * half;
    float s = bv;
    for (int k = 0; k < 64; ++k) {
      s += X1[(size_t)r * 64 + k] * W1[(size_t)k * 64 + col0 + lid];
      if (DUAL) s += X2[(size_t)r * 64 + k] * W2[(size_t)k * 64 + col0 + lid];
    }
    if (RELU) s = fmaxf(s, 0.0f);
    C[(size_t)r * 64 + col0 + lid] = s;
  }
#endif
}

// ---------------------------------------------------------------------------
// Elementwise / edge kernels
// ---------------------------------------------------------------------------
__global__ void zero_kernel(float* __restrict__ p, long long n) {
  long long i = (long long)blockIdx.x * blockDim.x + threadIdx.x;
  if (i < n) p[i] = 0.0f;
}

// Per-edge: e = ||pos[src]-pos[dst]||, deg[dst] += 1
__global__ void edge_prep_kernel(const int* __restrict__ src, const int* __restrict__ dst,
                                 const float* __restrict__ pos, float* __restrict__ e,
                                 float* __restrict__ deg, int ne) {
  int i = blockIdx.x * blockDim.x + threadIdx.x;
  if (i >= ne) return;
  int s = src[i], d = dst[i];
  float dx = pos[(size_t)s * 3 + 0] - pos[(size_t)d * 3 + 0];
  float dy = pos[(size_t)s * 3 + 1] - pos[(size_t)d * 3 + 1];
  float dz = pos[(size_t)s * 3 + 2] - pos[(size_t)d * 3 + 2];
  e[i] = sqrtf(dx * dx + dy * dy + dz * dz + 1e-12f);
  atomicAdd(deg + d, 1.0f);
}

__global__ void count_kernel(const int* __restrict__ batch, float* __restrict__ cnt, int n) {
  int i = blockIdx.x * blockDim.x + threadIdx.x;
  if (i < n) atomicAdd(cnt + batch[i], 1.0f);
}

__global__ void inv_kernel(float* __restrict__ x, int n) {
  int i = blockIdx.x * blockDim.x + threadIdx.x;
  if (i < n) x[i] = (x[i] > 0.0f) ? (1.0f / x[i]) : 0.0f;
}

// Wave-per-edge message + scatter: aggr[dst] += relu(A[dst] + B[src] + e*We)
// Each lane owns 2 of the 64 feature components (coalesced float2 from L2).
__global__ __launch_bounds__(256) void edge_msg_kernel(const int* __restrict__ src,
                                                       const int* __restrict__ dst,
                                                       const float* __restrict__ e,
                                                       const float* __restrict__ A,
                                                       const float* __restrict__ B,
                                                       const float* __restrict__ We,
                                                       float* __restrict__ aggr, int ne) {
  int gtid = blockIdx.x * blockDim.x + threadIdx.x;
  int eid = gtid >> 5;
  if (eid >= ne) return;
  int lane = threadIdx.x & 31;
  int s = src[eid], d = dst[eid];
  float ev = e[eid];
  float2 a = *(const float2*)(A + (size_t)d * 64 + lane * 2);
  float2 b = *(const float2*)(B + (size_t)s * 64 + lane * 2);
  float2 w = *(const float2*)(We + lane * 2);
  float m0 = fmaxf(a.x + b.x + ev * w.x, 0.0f);
  float m1 = fmaxf(a.y + b.y + ev * w.y, 0.0f);
  float* out = aggr + (size_t)d * 64 + lane * 2;
  atomicAdd(out, m0);
  atomicAdd(out + 1, m1);
}

// aggr[row][c] *= inv_deg[row]
__global__ void scale_rows_kernel(float* __restrict__ aggr, const float* __restrict__ invdeg,
                                  int n) {
  long long idx = (long long)blockIdx.x * blockDim.x + threadIdx.x;
  if (idx >= (long long)n * 64) return;
  int row = (int)(idx >> 6);
  aggr[idx] *= invdeg[row];
}

// p[batch[row]] += hs[row]  (segment sum over nodes into G graphs)
__global__ void pool_acc_kernel(const float* __restrict__ hs, const int* __restrict__ batch,
                                float* __restrict__ p, int n) {
  long long idx = (long long)blockIdx.x * blockDim.x + threadIdx.x;
  if (idx >= (long long)n * 64) return;
  int row = (int)(idx >> 6);
  int c = (int)(idx & 63);
  atomicAdd(p + (size_t)batch[row] * 64 + c, hs[idx]);
}

// pooled_s = relu((p * inv_cnt) @ pool_W + pool_b) -> msf[:, sOff:sOff+64]
__global__ void pool_small_kernel(const float* __restrict__ p, const float* __restrict__ invcnt,
                                  const float* __restrict__ W, const float* __restrict__ b,
                                  float* __restrict__ msf, int sOff) {
  int g = blockIdx.x;      // 0..G-1
  int j = threadIdx.x;     // 0..63
  float ic = invcnt[g];
  float s = b[j];
  for (int k = 0; k < 64; ++k)
    s += p[(size_t)g * 64 + k] * ic * W[(size_t)k * 64 + j];
  msf[(size_t)g * (SS * HH) + sOff + j] = fmaxf(s, 0.0f);
}

// out[g] = relu(msf[g] @ W1 + b1) @ W2 + b2   (192 -> 64 -> 1)
__global__ void head_kernel(const float* __restrict__ msf, const float* __restrict__ W1,
                            const float* __restrict__ b1, const float* __restrict__ W2,
                            const float* __restrict__ b2, float* __restrict__ out) {
  int g = blockIdx.x;
  int j = threadIdx.x;     // 0..63
  const float* row = msf + (size_t)g * (SS * HH);
  float s = b1[j];
  for (int k = 0; k < SS * HH; ++k) s += row[k] * W1[(size_t)k * 64 + j];
  float t = fmaxf(s, 0.0f) * W2[j];
  __shared__ float red[64];
  red[j] = t;
  __syncthreads();
  for (int off = 32; off > 0; off >>= 1) {
    if (j < off) red[j] += red[j + off];
    __syncthreads();
  }
  if (j == 0) out[g] = red[0] + b2[0];
}

// ---------------------------------------------------------------------------
extern "C" void kernel_launch(void* const* d_in, const int* in_sizes, int n_in,
                              void* d_out, int out_size, void* d_ws, size_t ws_size,
                              hipStream_t stream) {
  const float* x      = (const float*)d_in[0];
  const float* pos    = (const float*)d_in[1];
  const int*   batch  = (const int*)d_in[2];
  const int*   eidx   = (const int*)d_in[3];
  const float* enc_W  = (const float*)d_in[4];
  const float* enc_b  = (const float*)d_in[5];
  const float* msg_W  = (const float*)d_in[6];
  const float* msg_b  = (const float*)d_in[7];
  const float* upd_W  = (const float*)d_in[8];
  const float* upd_b  = (const float*)d_in[9];
  const float* pool_W = (const float*)d_in[10];
  const float* pool_b = (const float*)d_in[11];
  const float* geo_W1 = (const float*)d_in[12];
  const float* geo_b1 = (const float*)d_in[13];
  const float* geo_W2 = (const float*)d_in[14];
  const float* geo_b2 = (const float*)d_in[15];
  const float* fun_W1 = (const float*)d_in[16];
  const float* fun_b1 = (const float*)d_in[17];
  const float* fun_W2 = (const float*)d_in[18];
  const float* fun_b2 = (const float*)d_in[19];
  float* out = (float*)d_out;

  const int* esrc = eidx;
  const int* edst = eidx + EE;

  // Workspace layout (floats)
  float* ws = (float*)d_ws;
  const size_t NH = (size_t)NN * 64;
  float* h      = ws;             ws += NH;        // encoded node features
  float* A      = ws;             ws += NH;        // h@Wi + msg_b (per scale)
  float* B      = ws;             ws += NH;        // h@Wj ; later reused as hs
  float* aggr   = ws;             ws += NH;        // edge aggregation
  float* ebuf   = ws;             ws += EE;        // per-edge distance
  float* invdeg = ws;             ws += NN;        // degree -> 1/degree
  float* cnt    = ws;             ws += 64;        // graph counts -> 1/count
  float* p      = ws;             ws += 64 * 64;   // per-graph pooled sums
  float* msf    = ws;             ws += (size_t)GG * SS * HH;  // concat pooled

  const int TB = 256;
  const int gemmBlocks = (NN + 31) / 32;              // 2 row-tiles of 16 per block
  const long long nh = (long long)NN * 64;

  // --- degree / counts / encoder / edge scalars ---
  zero_kernel<<<(int)((NN + TB - 1) / TB), TB, 0, stream>>>(invdeg, NN);
  zero_kernel<<<1, 64, 0, stream>>>(cnt, 64);
  gemm64_wmma<true, false><<<gemmBlocks, TB, 0, stream>>>(x, enc_W, nullptr, nullptr,
                                                          enc_b, h, NN);
  edge_prep_kernel<<<(EE + TB - 1) / TB, TB, 0, stream>>>(esrc, edst, pos, ebuf, invdeg, EE);
  count_kernel<<<(NN + TB - 1) / TB, TB, 0, stream>>>(batch, cnt, NN);
  inv_kernel<<<(NN + TB - 1) / TB, TB, 0, stream>>>(invdeg, NN);
  inv_kernel<<<1, 64, 0, stream>>>(cnt, 64);

  // --- per-scale message passing ---
  for (int s = 0; s < SS; ++s) {
    const float* Wi = msg_W + (size_t)s * 129 * 64;
    const float* Wj = Wi + 64 * 64;
    const float* We = Wi + 128 * 64;
    // A = h@Wi + msg_b[s]   (bias folded in; relu happens per-edge)
    gemm64_wmma<false, false><<<gemmBlocks, TB, 0, stream>>>(h, Wi, nullptr, nullptr,
                                                             msg_b + (size_t)s * 64, A, NN);
    // B = h@Wj
    gemm64_wmma<false, false><<<gemmBlocks, TB, 0, stream>>>(h, Wj, nullptr, nullptr,
                                                             nullptr, B, NN);
    zero_kernel<<<(int)((nh + TB - 1) / TB), TB, 0, stream>>>(aggr, nh);
    // aggr[dst] += relu(A[dst] + B[src] + e*We)
    const long long ethreads = (long long)EE * 32;
    edge_msg_kernel<<<(int)((ethreads + TB - 1) / TB), TB, 0, stream>>>(esrc, edst, ebuf,
                                                                        A, B, We, aggr, EE);
    scale_rows_kernel<<<(int)((nh + TB - 1) / TB), TB, 0, stream>>>(aggr, invdeg, NN);
    // hs = relu(h @ updW1 + aggr @ updW2 + upd_b)   (write into B, A/B no longer needed)
    const float* Wu1 = upd_W + (size_t)s * 128 * 64;
    const float* Wu2 = Wu1 + 64 * 64;
    gemm64_wmma<true, true><<<gemmBlocks, TB, 0, stream>>>(h, Wu1, aggr, Wu2,
                                                           upd_b + (size_t)s * 64, B, NN);
    // p = segment_sum(hs, batch); pooled = relu(p*inv_cnt @ pool_W + pool_b)
    zero_kernel<<<16, 256, 0, stream>>>(p, 64 * 64);
    pool_acc_kernel<<<(int)((nh + TB - 1) / TB), TB, 0, stream>>>(B, batch, p, NN);
    pool_small_kernel<<<GG, 64, 0, stream>>>(p, cnt, pool_W + (size_t)s * 64 * 64,
                                             pool_b + (size_t)s * 64, msf, s * 64);
  }

  // --- heads: geom then func, concatenated in d_out ---
  head_kernel<<<GG, 64, 0, stream>>>(msf, geo_W1, geo_b1, geo_W2, geo_b2, out);
  head_kernel<<<GG, 64, 0, stream>>>(msf, fun_W1, fun_b1, fun_W2, fun_b2, out + GG);
}